// RoIAlignLayer_65901978190477
// MI455X (gfx1250) — compile-verified
//
#include <hip/hip_runtime.h>

// RoIAlign (torchvision, aligned=false) for MI455X / gfx1250.
// features: [N=2, C=256, H=200, W=200] f32 (NCHW), rois: [K=512, 5] f32
// out: [K, C, 7, 7] f32. scale=1.0, pooled=7, sampling_ratio=2.
//
// Memory-bound gather (AI ~2 flop/byte): no WMMA by design. Optimizations:
// block-uniform ROI -> SALU/SMEM ROI path, coalesced NT output stores,
// unrolled 16-load gather per output element, L2-warming prefetch.

#define POOLED 7
#define SR 2

namespace {
constexpr int C_ = 256;
constexpr int H_ = 200;
constexpr int W_ = 200;
constexpr int BINS = POOLED * POOLED;          // 49
constexpr int BLOCK = 256;                     // 8 wave32
// C_*BINS = 12544 = 49 * 256  ->  exactly 49 blocks per ROI, k uniform per block
constexpr int BLOCKS_PER_ROI = (C_ * BINS) / BLOCK; // 49
} // namespace

__global__ __launch_bounds__(BLOCK) void roi_align_gfx1250(
    const float* __restrict__ feat,   // [N, C, H, W]
    const float* __restrict__ rois,   // [K, 5]
    float* __restrict__ out)          // [K, C, 7, 7]
{
    // k is workgroup-uniform -> ROI loads/math go scalar (s_load / SALU).
    const int k     = blockIdx.x / BLOCKS_PER_ROI;
    const int local = (blockIdx.x % BLOCKS_PER_ROI) * BLOCK + threadIdx.x; // 0..12543
    const int c     = local / BINS;
    const int bin   = local - c * BINS;
    const int py    = bin / POOLED;
    const int px    = bin - py * POOLED;

    const float* __restrict__ r = rois + (size_t)k * 5;
    const int   b  = (int)r[0];
    const float x1 = r[1];
    const float y1 = r[2];
    const float roi_w = fmaxf(r[3] - x1, 1.0f);
    const float roi_h = fmaxf(r[4] - y1, 1.0f);
    const float bin_w = roi_w * (1.0f / POOLED);
    const float bin_h = roi_h * (1.0f / POOLED);

    const float* __restrict__ fplane =
        feat + ((size_t)b * C_ + (size_t)c) * (size_t)(H_ * W_);

    // Warm L2/TLB for the first sample's two rows before the dependent gathers.
    {
        const float yp = y1 + ((float)py + 0.25f) * bin_h;
        const float xp = x1 + ((float)px + 0.25f) * bin_w;
        const float ycp = fminf(fmaxf(yp, 0.0f), (float)(H_ - 1));
        const float xcp = fminf(fmaxf(xp, 0.0f), (float)(W_ - 1));
        int y0p = (int)floorf(ycp); y0p = y0p < 0 ? 0 : (y0p > H_ - 2 ? H_ - 2 : y0p);
        int x0p = (int)floorf(xcp); x0p = x0p < 0 ? 0 : (x0p > W_ - 2 ? W_ - 2 : x0p);
        const float* p = fplane + (size_t)y0p * W_ + x0p;
        __builtin_prefetch(p, 0, 3);           // global_prefetch_b8 (row y0)
        __builtin_prefetch(p + W_, 0, 3);      // global_prefetch_b8 (row y0+1)
    }

    float acc = 0.0f;
#pragma unroll
    for (int iy = 0; iy < SR; ++iy) {
        const float y = y1 + ((float)py + ((float)iy + 0.5f) * (1.0f / SR)) * bin_h;
        const float yc = fminf(fmaxf(y, 0.0f), (float)(H_ - 1));
        int y0 = (int)floorf(yc);
        y0 = y0 < 0 ? 0 : (y0 > H_ - 2 ? H_ - 2 : y0);
        const float ly = yc - (float)y0;
        const float hy = 1.0f - ly;
        const bool yok = (y >= -1.0f) && (y <= (float)H_);
        const float* __restrict__ row = fplane + (size_t)y0 * W_;

#pragma unroll
        for (int ix = 0; ix < SR; ++ix) {
            const float x = x1 + ((float)px + ((float)ix + 0.5f) * (1.0f / SR)) * bin_w;
            const float xc = fminf(fmaxf(x, 0.0f), (float)(W_ - 1));
            int x0 = (int)floorf(xc);
            x0 = x0 < 0 ? 0 : (x0 > W_ - 2 ? W_ - 2 : x0);
            const float lx = xc - (float)x0;
            const float hx = 1.0f - lx;
            const bool ok = yok && (x >= -1.0f) && (x <= (float)W_);

            const float v00 = row[x0];
            const float v01 = row[x0 + 1];
            const float v10 = row[x0 + W_];
            const float v11 = row[x0 + W_ + 1];

            const float v = hy * fmaf(hx, v00, lx * v01)
                          + ly * fmaf(hx, v10, lx * v11);
            acc += ok ? v : 0.0f;
        }
    }

    // Output index == blockIdx.x*BLOCK + threadIdx.x (fully coalesced).
    const size_t oidx = (size_t)blockIdx.x * BLOCK + threadIdx.x;
    // Write-once output: non-temporal store keeps feature map resident in L2.
    __builtin_nontemporal_store(acc * (1.0f / (SR * SR)), out + oidx);
}

extern "C" void kernel_launch(void* const* d_in, const int* in_sizes, int n_in,
                              void* d_out, int out_size, void* d_ws, size_t ws_size,
                              hipStream_t stream) {
    (void)n_in; (void)d_ws; (void)ws_size; (void)out_size;
    const float* feat = (const float*)d_in[0];
    const float* rois = (const float*)d_in[1];
    float* out        = (float*)d_out;

    const int K = in_sizes[1] / 5;                  // 512
    const int grid = K * BLOCKS_PER_ROI;            // 512 * 49 = 25088 blocks
    roi_align_gfx1250<<<grid, BLOCK, 0, stream>>>(feat, rois, out);
}